// Net_33157147525934
// MI455X (gfx1250) — compile-verified
//
#include <hip/hip_runtime.h>
#include <float.h>
#include <math.h>

#define HID 128
#define MIN_SCORE 0.1f
#define TOLV 1e-7f
#define BN_EPS 1e-5f

typedef float v2f __attribute__((ext_vector_type(2)));
typedef float v8f __attribute__((ext_vector_type(8)));

// ---------------- small helpers ----------------

__global__ void k_init_cnt(int* cnt, int B) {
    int i = threadIdx.x;
    if (i < B) cnt[i] = 0;
}

__global__ void k_count(const int* __restrict__ batch, int* cnt, int N) {
    int n = blockIdx.x * 256 + threadIdx.x;
    if (n < N) atomicAdd(&cnt[batch[n]], 1);
}

__global__ void k_scan(const int* __restrict__ cnt, int* off, int B) {
    if (threadIdx.x == 0) {
        int a = 0;
        for (int b = 0; b < B; ++b) { off[b] = a; a += cnt[b]; }
        off[B] = a;
    }
}

__global__ void k_logits(const float* __restrict__ x, const float* __restrict__ w,
                         float* __restrict__ logits, int N) {
    int n = blockIdx.x * 256 + threadIdx.x;
    if (n < N) logits[n] = x[2 * n] * w[0] + x[2 * n + 1] * w[1];
}

// per-graph max + sum(exp) : one block per graph, fixed-order reduction (deterministic)
__global__ void k_seg_stats(const float* __restrict__ logits, const int* __restrict__ off,
                            float* gmax, float* gsum) {
    __shared__ float red[256];
    const int b = blockIdx.x;
    const int lo = off[b], hi = off[b + 1];
    float m = -FLT_MAX;
    for (int n = lo + threadIdx.x; n < hi; n += 256) m = fmaxf(m, logits[n]);
    red[threadIdx.x] = m; __syncthreads();
    for (int s = 128; s > 0; s >>= 1) {
        if (threadIdx.x < s) red[threadIdx.x] = fmaxf(red[threadIdx.x], red[threadIdx.x + s]);
        __syncthreads();
    }
    m = red[0]; __syncthreads();
    float acc = 0.0f;
    for (int n = lo + threadIdx.x; n < hi; n += 256) acc += expf(logits[n] - m);
    red[threadIdx.x] = acc; __syncthreads();
    for (int s = 128; s > 0; s >>= 1) {
        if (threadIdx.x < s) red[threadIdx.x] += red[threadIdx.x + s];
        __syncthreads();
    }
    if (threadIdx.x == 0) { gmax[b] = m; gsum[b] = red[0]; }
}

// score, keep mask, scaled input features; also zeroes deg
__global__ void k_score_keep(const float* __restrict__ x, const int* __restrict__ batch,
                             const float* __restrict__ logits,
                             const float* __restrict__ gmax, const float* __restrict__ gsum,
                             float* kf, float* hx, float* deg, int N) {
    int n = blockIdx.x * 256 + threadIdx.x;
    if (n >= N) return;
    int b = batch[n];
    float s = expf(logits[n] - gmax[b]) / gsum[b];
    float smax = 1.0f / gsum[b] - TOLV;          // max score in graph == exp(0)/sum
    float thr = fminf(smax, MIN_SCORE);
    float k = (s > thr) ? 1.0f : 0.0f;
    kf[n] = k;
    float sc = s * k;
    hx[2 * n]     = x[2 * n]     * sc;
    hx[2 * n + 1] = x[2 * n + 1] * sc;
    deg[n] = 0.0f;
}

// degree: sums of exact 0.0/1.0 -> deterministic float atomics
__global__ void k_degree(const int* __restrict__ src, const int* __restrict__ dst,
                         const float* __restrict__ kf, float* deg, int E, int N) {
    int i = blockIdx.x * 256 + threadIdx.x;
    if (i < E) {
        int s = src[i], d = dst[i];
        float w = kf[s] * kf[d];
        if (w != 0.0f) atomicAdd(&deg[d], 1.0f);
    } else if (i < E + N) {
        int n = i - E;
        if (kf[n] != 0.0f) atomicAdd(&deg[n], 1.0f);   // self loop
    }
}

__global__ void k_dinv(float* deg, int N) {
    int n = blockIdx.x * 256 + threadIdx.x;
    if (n >= N) return;
    float d = deg[n];
    deg[n] = (d > 0.0f) ? (1.0f / sqrtf(d)) : 0.0f;
}

// layer-1 linear (K=2): trivially memory-bound, no WMMA needed
__global__ void k_lin1(const float* __restrict__ hx, const float* __restrict__ W1,
                       float* __restrict__ xw, int N) {
    int t = blockIdx.x * 256 + threadIdx.x;
    if (t >= N * HID) return;
    int n = t >> 7, j = t & 127;
    xw[t] = hx[2 * n] * W1[j] + hx[2 * n + 1] * W1[HID + j];
}

__global__ void k_zero4(float4* p, int n4) {
    int i = blockIdx.x * 256 + threadIdx.x;
    if (i < n4) p[i] = make_float4(0.f, 0.f, 0.f, 0.f);
}

// ---------------- exact-fp32 WMMA GEMM: Out[N x 128] = A[N x 128] @ W[128 x 128] ----------------
// Block: 256 threads (8 waves); block computes a 128-row stripe x all 128 cols.
// K sliced in 16-wide LDS slabs. Fragments are single aligned b64 LDS loads:
//   sA stride 20 : 16B-aligned rows (float4 staging), conflict-free frag reads
//   sW4[kg][col][4] : K-quad per column contiguous -> B frag = one ds_load_b64
__global__ __launch_bounds__(256)
void k_gemm128(const float* __restrict__ A, const float* __restrict__ W,
               float* __restrict__ Out, int nrows) {
    __shared__ __align__(16) float sA[128][20];
    __shared__ __align__(16) float sW4[4][HID][4];

    const int tid  = threadIdx.x;
    const int wave = tid >> 5;        // 0..7 : row tile within block
    const int lane = tid & 31;
    const int m    = lane & 15;       // row (A) / col (B,D) within tile
    const int half = lane >> 4;       // K-half select per ISA f32 layout
    const int row0 = blockIdx.x * 128;

    v8f acc[8];
#pragma unroll
    for (int t = 0; t < 8; ++t) { v8f z = {}; acc[t] = z; }

    for (int ks = 0; ks < HID; ks += 16) {
        // stage A slab: 128 rows x 16 K, float4 per thread x2 (coalesced, b128 LDS stores)
        for (int i4 = tid; i4 < 128 * 4; i4 += 256) {
            int r = i4 >> 2, c4 = (i4 & 3) << 2;
            int gr = row0 + r;
            float4 v = (gr < nrows) ? *(const float4*)(A + (size_t)gr * HID + ks + c4)
                                    : make_float4(0.f, 0.f, 0.f, 0.f);
            *(float4*)&sA[r][c4] = v;
        }
        // stage W slab repacked: (ks+r, c) -> sW4[r>>2][c][r&3]
        for (int i = tid; i < 16 * HID; i += 256) {
            int r = i >> 7, c = i & 127;
            sW4[r >> 2][c][r & 3] = W[(size_t)(ks + r) * HID + c];
        }
        __syncthreads();

        const int arow = wave * 16 + m;
#pragma unroll
        for (int kg = 0; kg < 4; ++kg) {
            const int kk = kg * 4 + half * 2;
            // A 16x4 f32: lane(m,half): VGPR0=K(2*half), VGPR1=K(2*half+1)
            v2f afrag = *(const v2f*)&sA[arow][kk];
#pragma unroll
            for (int ct = 0; ct < 8; ++ct) {
                // B 4x16 f32 mirrored layout: one aligned b64 load, no repack movs
                v2f bfrag = *(const v2f*)&sW4[kg][ct * 16 + m][half * 2];
                acc[ct] = __builtin_amdgcn_wmma_f32_16x16x4_f32(
                    false, afrag, false, bfrag, (short)0, acc[ct], false, false);
            }
        }
        __syncthreads();
    }

    // D 16x16 f32: VGPR r holds row M = half*8 + r, col = lane&15
#pragma unroll
    for (int ct = 0; ct < 8; ++ct) {
#pragma unroll
        for (int r = 0; r < 8; ++r) {
            int row = row0 + wave * 16 + half * 8 + r;
            if (row < nrows) Out[(size_t)row * HID + ct * 16 + m] = acc[ct][r];
        }
    }
}

// ---------------- edge scatter: one wave32 per edge, float4 per lane ----------------
__global__ __launch_bounds__(256)
void k_aggregate(const int* __restrict__ src, const int* __restrict__ dst,
                 const float* __restrict__ kf, const float* __restrict__ dinv,
                 const float* __restrict__ xw, float* __restrict__ agg, int E, int N) {
    int g = blockIdx.x * 256 + threadIdx.x;
    int w = g >> 5, lane = g & 31;
    if (w >= E + N) return;
    int s, d; float coef;
    if (w < E) {
        s = src[w]; d = dst[w];
        float ew = kf[s] * kf[d];
        if (ew == 0.0f) return;
        coef = dinv[s] * dinv[d];
    } else {
        s = d = w - E;                               // self loop
        if (kf[s] == 0.0f) return;
        float di = dinv[s];
        coef = di * di;
    }
    const float4 v = ((const float4*)(xw + (size_t)s * HID))[lane];
    float* o = agg + (size_t)d * HID + lane * 4;
    atomicAdd(o + 0, coef * v.x);
    atomicAdd(o + 1, coef * v.y);
    atomicAdd(o + 2, coef * v.z);
    atomicAdd(o + 3, coef * v.w);
}

// bias + relu + eval-mode BN, in place
__global__ void k_post(float* __restrict__ agg, const float* __restrict__ b,
                       const float* __restrict__ g, const float* __restrict__ be,
                       const float* __restrict__ rm, const float* __restrict__ rv, int N) {
    int t = blockIdx.x * 256 + threadIdx.x;
    if (t >= N * HID) return;
    int j = t & 127;
    float v = agg[t] + b[j];
    v = fmaxf(v, 0.0f);
    v = (v - rm[j]) * rsqrtf(rv[j] + BN_EPS) * g[j] + be[j];
    agg[t] = v;
}

// masked per-graph max pool (deterministic sequential scan per (graph, channel))
__global__ void k_pool(const float* __restrict__ h, const float* __restrict__ kf,
                       const int* __restrict__ off, float* __restrict__ pooled) {
    const int b = blockIdx.x, j = threadIdx.x;      // 128 threads
    float m = -FLT_MAX;
    const int lo = off[b], hi = off[b + 1];
    for (int n = lo; n < hi; ++n)
        if (kf[n] != 0.0f) m = fmaxf(m, h[(size_t)n * HID + j]);
    pooled[b * HID + j] = m;
}

__global__ void k_head(const float* __restrict__ pooled, const float* __restrict__ linW,
                       const float* __restrict__ linb, float* __restrict__ out, int B) {
    int b = blockIdx.x * blockDim.x + threadIdx.x;
    if (b >= B) return;
    float l[3];
    for (int c = 0; c < 3; ++c) {
        float a = linb[c];
        for (int k = 0; k < HID; ++k) a += pooled[b * HID + k] * linW[k * 3 + c];
        l[c] = a;
    }
    float m = fmaxf(l[0], fmaxf(l[1], l[2]));
    float s = expf(l[0] - m) + expf(l[1] - m) + expf(l[2] - m);
    float lse = m + logf(s);
    for (int c = 0; c < 3; ++c) out[b * 3 + c] = l[c] - lse;
}

// ---------------- launch ----------------

extern "C" void kernel_launch(void* const* d_in, const int* in_sizes, int n_in,
                              void* d_out, int out_size, void* d_ws, size_t ws_size,
                              hipStream_t stream) {
    const int N = in_sizes[0] / 2;
    const int E = in_sizes[1] / 2;
    const int B = out_size / 3;

    const float* x     = (const float*)d_in[0];
    const int*   ei    = (const int*)d_in[1];     // [0..E) = src, [E..2E) = dst
    const int*   batch = (const int*)d_in[2];
    const float* topkw = (const float*)d_in[3];
    const float* W[3]  = { (const float*)d_in[4],  (const float*)d_in[10], (const float*)d_in[16] };
    const float* bb[3] = { (const float*)d_in[5],  (const float*)d_in[11], (const float*)d_in[17] };
    const float* gg[3] = { (const float*)d_in[6],  (const float*)d_in[12], (const float*)d_in[18] };
    const float* bebe[3]={ (const float*)d_in[7],  (const float*)d_in[13], (const float*)d_in[19] };
    const float* rm[3] = { (const float*)d_in[8],  (const float*)d_in[14], (const float*)d_in[20] };
    const float* rv[3] = { (const float*)d_in[9],  (const float*)d_in[15], (const float*)d_in[21] };
    const float* linW  = (const float*)d_in[22];
    const float* linb  = (const float*)d_in[23];

    char* ws = (char*)d_ws;
    auto alloc = [&](size_t bytes) -> void* {
        void* p = (void*)ws;
        ws += (bytes + 255) & ~(size_t)255;
        return p;
    };
    float* logits = (float*)alloc((size_t)N * 4);
    float* kf     = (float*)alloc((size_t)N * 4);
    float* hx     = (float*)alloc((size_t)N * 2 * 4);
    float* dinv   = (float*)alloc((size_t)N * 4);        // deg -> dinv in place
    int*   cnt    = (int*)alloc((size_t)B * 4);
    int*   off    = (int*)alloc((size_t)(B + 1) * 4);
    float* gmax   = (float*)alloc((size_t)B * 4);
    float* gsum   = (float*)alloc((size_t)B * 4);
    float* pooled = (float*)alloc((size_t)B * HID * 4);
    float* bufA   = (float*)alloc((size_t)N * HID * 4);  // xw
    float* bufC   = (float*)alloc((size_t)N * HID * 4);  // agg / h (alternates in place)

    const int TB = 256;
    const int gN   = (N + TB - 1) / TB;
    const int gNH  = (N * HID + TB - 1) / TB;
    const int gNH4 = (N * HID / 4 + TB - 1) / TB;
    const int gEN  = (E + N + TB - 1) / TB;
    const long long aggThreads = (long long)(E + N) * 32;
    const int gAGG = (int)((aggThreads + TB - 1) / TB);
    const int gGEMM = (N + 127) / 128;

    // --- TopK pooling (min_score mode) ---
    k_init_cnt<<<1, 256, 0, stream>>>(cnt, B);
    k_count<<<gN, TB, 0, stream>>>(batch, cnt, N);
    k_scan<<<1, 64, 0, stream>>>(cnt, off, B);
    k_logits<<<gN, TB, 0, stream>>>(x, topkw, logits, N);
    k_seg_stats<<<B, 256, 0, stream>>>(logits, off, gmax, gsum);
    k_score_keep<<<gN, TB, 0, stream>>>(x, batch, logits, gmax, gsum, kf, hx, dinv, N);

    // --- symmetric normalization ---
    k_degree<<<gEN, TB, 0, stream>>>(ei, ei + E, kf, dinv, E, N);
    k_dinv<<<gN, TB, 0, stream>>>(dinv, N);

    // --- 3 GCN layers ---
    for (int l = 0; l < 3; ++l) {
        if (l == 0)
            k_lin1<<<gNH, TB, 0, stream>>>(hx, W[0], bufA, N);
        else
            k_gemm128<<<gGEMM, 256, 0, stream>>>(bufC, W[l], bufA, N);
        k_zero4<<<gNH4, TB, 0, stream>>>((float4*)bufC, N * HID / 4);
        k_aggregate<<<gAGG, TB, 0, stream>>>(ei, ei + E, kf, dinv, bufA, bufC, E, N);
        k_post<<<gNH, TB, 0, stream>>>(bufC, bb[l], gg[l], bebe[l], rm[l], rv[l], N);
    }

    // --- readout ---
    k_pool<<<B, HID, 0, stream>>>(bufC, kf, off, pooled);
    k_head<<<1, 64, 0, stream>>>(pooled, linW, linb, (float*)d_out, B);
}